// VentralModel_77790447665762
// MI455X (gfx1250) — compile-verified
//
#include <hip/hip_runtime.h>

typedef __attribute__((ext_vector_type(2))) float v2f;
typedef __attribute__((ext_vector_type(4))) float v4f;
typedef __attribute__((ext_vector_type(8))) float v8f;

#define N_WIN 300
#define KS0 (512*512)
#define KS1 (256*256)
#define KS2 (128*128)
#define KS3 (64*64)

// workspace float offsets: modulus rows, K-contiguous per column
// scale0: cols 0..3 = orientations, col 4 = image row
#define M0_OFF 0
#define M1_OFF (5*KS0)
#define M2_OFF (M1_OFF + 4*KS1)
#define M3_OFF (M2_OFF + 4*KS2)
#define M_TOTAL (M3_OFF + 4*KS3)          // 1,654,784 floats (~6.6 MB)

#define OUT_SIZE 5100
#define CHUNK 1024
#define ZT 1e-20f

__global__ void zero_kernel(float* __restrict__ out, int* __restrict__ wnp) {
    int i = blockIdx.x * blockDim.x + threadIdx.x;
    if (i < OUT_SIZE) out[i] = 0.0f;
    if (i < 4 * N_WIN) wnp[i] = 0;
}

__global__ void prep_kernel(const float* __restrict__ image,
                            const float* __restrict__ pyr0,
                            const float* __restrict__ pyr1,
                            const float* __restrict__ pyr2,
                            const float* __restrict__ pyr3,
                            float* __restrict__ m) {
    const long total = M_TOTAL;
    for (long i = (long)blockIdx.x * blockDim.x + threadIdx.x; i < total;
         i += (long)gridDim.x * blockDim.x) {
        float v;
        if (i < 4L * KS0) {                       // scale0 modulus, (o*K + k)
            float re = pyr0[2 * i], im = pyr0[2 * i + 1];
            v = sqrtf(re * re + im * im);
        } else if (i < 5L * KS0) {                // image row (col 4 of scale0)
            v = image[i - 4L * KS0];
        } else if (i < (long)M2_OFF) {
            long r = i - M1_OFF;
            float re = pyr1[2 * r], im = pyr1[2 * r + 1];
            v = sqrtf(re * re + im * im);
        } else if (i < (long)M3_OFF) {
            long r = i - M2_OFF;
            float re = pyr2[2 * r], im = pyr2[2 * r + 1];
            v = sqrtf(re * re + im * im);
        } else {
            long r = i - M3_OFF;
            float re = pyr3[2 * r], im = pyr3[2 * r + 1];
            v = sqrtf(re * re + im * im);
        }
        m[i] = v;
    }
}

// One wave per block. Tile of 16 windows (A rows) x 16 columns (B): cols
// 0..3 orientations, col 4 = image (scale 0 only). Branchless: out-of-range
// rows/cols are clamped (their C entries are discarded at writeback).
// Each lane streams b128 (8 k per iteration -> 2 WMMAs), software-pipelined.
// A (window) loads are non-temporal: 418 MB streamed once, keep L2 for B.
__global__ __launch_bounds__(32) void pool_wmma_kernel(
    const float* __restrict__ win,     // (N_WIN, K) row-major
    const float* __restrict__ mcols,   // (ncols, K) row-major
    float* __restrict__ out,
    int* __restrict__ wnp,             // per-window nonzero counts (this scale)
    int K, int ncols, int out_base, int lum_base /* -1 if none */) {
    const int lane = threadIdx.x;           // 0..31
    const int hl   = lane >> 4;             // lane half
    const int wl   = lane & 15;             // A row index / B column index
    const int tile = blockIdx.x;
    const int k0   = blockIdx.y * CHUNK;

    const int w    = tile * 16 + wl;
    const int wcl  = (w  < N_WIN) ? w  : (N_WIN - 1);   // clamped A row
    const int ncl  = (wl < ncols) ? wl : (ncols - 1);   // clamped B row

    const float* ap = win   + (size_t)wcl * K + k0 + 4 * hl;
    const float* bp = mcols + (size_t)ncl * K + k0 + 4 * hl;

    v8f c = {0.f, 0.f, 0.f, 0.f, 0.f, 0.f, 0.f, 0.f};
    int cnt = 0;

    v4f a = __builtin_nontemporal_load((const v4f*)ap);   // TH=NT stream
    v4f b = *(const v4f*)bp;                              // TH=RT, L2-hot

    const int iters = CHUNK / 8;            // 128
    #pragma unroll 4
    for (int it = 1; it < iters; ++it) {
        v4f an = __builtin_nontemporal_load((const v4f*)(ap + it * 8));
        v4f bn = *(const v4f*)(bp + it * 8);
        cnt += (a.x > ZT) + (a.y > ZT) + (a.z > ZT) + (a.w > ZT);
        v2f alo = __builtin_shufflevector(a, a, 0, 1);
        v2f ahi = __builtin_shufflevector(a, a, 2, 3);
        v2f blo = __builtin_shufflevector(b, b, 0, 1);
        v2f bhi = __builtin_shufflevector(b, b, 2, 3);
        c = __builtin_amdgcn_wmma_f32_16x16x4_f32(false, alo, false, blo,
                                                  (short)0, c, false, false);
        c = __builtin_amdgcn_wmma_f32_16x16x4_f32(false, ahi, false, bhi,
                                                  (short)0, c, false, false);
        a = an;
        b = bn;
    }
    // epilogue
    cnt += (a.x > ZT) + (a.y > ZT) + (a.z > ZT) + (a.w > ZT);
    {
        v2f alo = __builtin_shufflevector(a, a, 0, 1);
        v2f ahi = __builtin_shufflevector(a, a, 2, 3);
        v2f blo = __builtin_shufflevector(b, b, 0, 1);
        v2f bhi = __builtin_shufflevector(b, b, 2, 3);
        c = __builtin_amdgcn_wmma_f32_16x16x4_f32(false, alo, false, blo,
                                                  (short)0, c, false, false);
        c = __builtin_amdgcn_wmma_f32_16x16x4_f32(false, ahi, false, bhi,
                                                  (short)0, c, false, false);
    }

    if (w < N_WIN) atomicAdd(&wnp[w], cnt);

    // C/D 16x16 f32 layout: VGPR j, lanes<16 -> row M=j, lanes>=16 -> row M=j+8
    #pragma unroll
    for (int j = 0; j < 8; ++j) {
        int wrow = tile * 16 + j + 8 * hl;
        if (wrow < N_WIN) {
            float val = c[j];
            if (wl < 4)
                atomicAdd(&out[out_base + wl * N_WIN + wrow], val);
            else if (wl == 4 && lum_base >= 0)
                atomicAdd(&out[lum_base + wrow], val);
        }
    }
}

__global__ void finalize_kernel(float* __restrict__ out,
                                const int* __restrict__ wnp) {
    int i = blockIdx.x * blockDim.x + threadIdx.x;
    if (i >= OUT_SIZE) return;
    int s, w;
    if (i < 4800) {
        s = i / 1200;
        w = (i % 1200) % N_WIN;
    } else {
        s = 0;                  // mean_lum divides by win0 counts
        w = i - 4800;
    }
    out[i] /= (float)wnp[s * N_WIN + w];
}

extern "C" void kernel_launch(void* const* d_in, const int* in_sizes, int n_in,
                              void* d_out, int out_size, void* d_ws, size_t ws_size,
                              hipStream_t stream) {
    // setup_inputs order: image, pyr0, win0, pyr1, win1, pyr2, win2, pyr3, win3
    const float* image = (const float*)d_in[0];
    const float* pyr0  = (const float*)d_in[1];
    const float* win0  = (const float*)d_in[2];
    const float* pyr1  = (const float*)d_in[3];
    const float* win1  = (const float*)d_in[4];
    const float* pyr2  = (const float*)d_in[5];
    const float* win2  = (const float*)d_in[6];
    const float* pyr3  = (const float*)d_in[7];
    const float* win3  = (const float*)d_in[8];
    (void)in_sizes; (void)n_in; (void)out_size; (void)ws_size;

    float* out = (float*)d_out;
    float* m   = (float*)d_ws;                                  // modulus rows
    int*   wnp = (int*)((char*)d_ws + (size_t)M_TOTAL * 4);     // 4*300 counts

    zero_kernel<<<(OUT_SIZE + 255) / 256, 256, 0, stream>>>(out, wnp);
    prep_kernel<<<2048, 256, 0, stream>>>(image, pyr0, pyr1, pyr2, pyr3, m);

    const int tiles = (N_WIN + 15) / 16;   // 19
    pool_wmma_kernel<<<dim3(tiles, KS0 / CHUNK), 32, 0, stream>>>(
        win0, m + M0_OFF, out, wnp + 0,     KS0, 5, 0,    4800);
    pool_wmma_kernel<<<dim3(tiles, KS1 / CHUNK), 32, 0, stream>>>(
        win1, m + M1_OFF, out, wnp + 300,   KS1, 4, 1200, -1);
    pool_wmma_kernel<<<dim3(tiles, KS2 / CHUNK), 32, 0, stream>>>(
        win2, m + M2_OFF, out, wnp + 600,   KS2, 4, 2400, -1);
    pool_wmma_kernel<<<dim3(tiles, KS3 / CHUNK), 32, 0, stream>>>(
        win3, m + M3_OFF, out, wnp + 900,   KS3, 4, 3600, -1);

    finalize_kernel<<<(OUT_SIZE + 255) / 256, 256, 0, stream>>>(out, wnp);
}